// NewRayAttentionLayer_27788438405628
// MI455X (gfx1250) — compile-verified
//
#include <hip/hip_runtime.h>
#include <math.h>

// Problem constants (from the reference)
#define NN   2
#define LL   384
#define HH   12
#define DQKc 16
#define DVc  16
#define NQKc 4
#define NVVc 8
#define HIDc 128
#define FEATc 576          // H*(DV + NVV*4)
#define EPSc 1e-6f

typedef __attribute__((ext_vector_type(2))) float v2f;
typedef __attribute__((ext_vector_type(8))) float v8f;

// Full-precision CDNA5 matrix op: D = A(16x4,f32) * B(4x16,f32) + C(16x16,f32)
static __device__ __forceinline__ v8f wmma_f32x4(v2f a, v2f b, v8f c) {
  return __builtin_amdgcn_wmma_f32_16x16x4_f32(
      /*neg_a=*/false, a, /*neg_b=*/false, b,
      /*c_mod=*/(short)0, c, /*reuse_a=*/false, /*reuse_b=*/false);
}

// ---------------------------------------------------------------------------
// Generic fp32 WMMA GEMM: C[M,Nc] = A[M,K] @ B[K,Nc] (+ bias[Nc]).
// A row-major, B row-major. One wave computes one 16x16 C tile.
// Out-of-range columns are CLAMPED on load (uniform control flow, no EXEC
// divergence in the inner loop) and masked only on the store.
// Fragment layouts per CDNA5 ISA 7.12.2:
//   A frag (2 VGPR):  elem r -> A[m0 + (lane&15)][k0 + r + 2*(lane>>4)]
//   B frag (2 VGPR):  elem r -> B[k0 + r + 2*(lane>>4)][n0 + (lane&15)]
//   C frag (8 VGPR):  elem r -> C[m0 + r + 8*(lane>>4)][n0 + (lane&15)]
// ---------------------------------------------------------------------------
__global__ __launch_bounds__(32)
void nra_gemm_bias_kernel(const float* __restrict__ A, const float* __restrict__ B,
                          const float* __restrict__ bias, float* __restrict__ C,
                          int M, int Nc, int K) {
  const int lane = threadIdx.x;
  const int lm = lane & 15, half = lane >> 4;
  const int m0 = blockIdx.y * 16, n0 = blockIdx.x * 16;
  const int arow = m0 + lm;
  const int col  = n0 + lm;
  const bool cok = (col < Nc);
  const int bcol = cok ? col : (Nc - 1);   // clamp: loads always in-bounds

  const float* Ap = A + (size_t)arow * K;
  const float* Bp = B + bcol;

  v8f acc = {};
  for (int k0 = 0; k0 < K; k0 += 4) {
    const int ka = k0 + 2 * half;
    v2f af, bf;
    af.x = Ap[ka];
    af.y = Ap[ka + 1];
    bf.x = Bp[(size_t)(ka)     * Nc];
    bf.y = Bp[(size_t)(ka + 1) * Nc];
    acc = wmma_f32x4(af, bf, acc);
  }

  if (cok) {
    const float bv = (bias != nullptr) ? bias[col] : 0.0f;
#pragma unroll
    for (int r = 0; r < 8; ++r) {
      const int mrow = m0 + r + 8 * half;
      C[(size_t)mrow * Nc + col] = acc[r] + bv;
    }
  }
}

// ---------------------------------------------------------------------------
// Apply global frame in-place:  p <- R[n,l] * p + t[n,l]
// for b (H vectors/token), kv (H*NQK), vv (H*NVV).
// ---------------------------------------------------------------------------
__global__ void nra_frame_kernel(const float* __restrict__ R, const float* __restrict__ t,
                                 float* __restrict__ bg, float* __restrict__ kvg,
                                 float* __restrict__ vvg) {
  const int NB = NN * LL * HH;          // 9216
  const int NK = NB * NQKc;             // 36864
  const int NV = NB * NVVc;             // 73728
  int idx = blockIdx.x * blockDim.x + threadIdx.x;
  float* p; int tok;
  if (idx < NB)            { p = bg  + (size_t)idx * 3;            tok = idx / HH; }
  else if (idx < NB + NK)  { int j = idx - NB;      p = kvg + (size_t)j * 3; tok = j / (HH * NQKc); }
  else if (idx < NB + NK + NV) { int j = idx - NB - NK; p = vvg + (size_t)j * 3; tok = j / (HH * NVVc); }
  else return;
  const float* Rm = R + (size_t)tok * 9;
  const float* tv = t + (size_t)tok * 3;
  const float x = p[0], y = p[1], z = p[2];
  p[0] = Rm[0] * x + Rm[1] * y + Rm[2] * z + tv[0];
  p[1] = Rm[3] * x + Rm[4] * y + Rm[5] * z + tv[1];
  p[2] = Rm[6] * x + Rm[7] * y + Rm[8] * z + tv[2];
}

// ---------------------------------------------------------------------------
// Fused attention: one block = (n, h, 16-row i-tile). 256 threads = 8 waves.
//  phase 1: S = (q k^T) * DQK^-0.5           (WMMA, 8 waves x 3 j-tiles)
//  phase 2: S += ray term; diag/mask         (VALU; acosf/sqrtf)
//  phase 3: row softmax                      (wave32 shfl reductions)
//  phase 4: [o | ov_g] = S @ [v | vv_g]      (WMMA, 3 waves, K=384)
//  phase 5: ov = Rinv (ov_g - t), norms -> feats
// ---------------------------------------------------------------------------
__global__ __launch_bounds__(256)
void nra_attn_kernel(const float* __restrict__ q, const float* __restrict__ k,
                     const float* __restrict__ v, const float* __restrict__ a,
                     const float* __restrict__ bg, const float* __restrict__ kvg,
                     const float* __restrict__ vvg, const float* __restrict__ R,
                     const float* __restrict__ t, const unsigned char* __restrict__ mask,
                     const float* __restrict__ alpha, const float* __restrict__ beta,
                     float* __restrict__ feats) {
  __shared__ float S[16][LL + 4];   // attention row block
  __shared__ float Si[16][32];      // per-row: Rinv(9) bg(3) t(3) a(12) |a|(4)
  __shared__ float Ot[16][48];      // [o(16) | ov_g(24) | pad(8)]

  const int b  = blockIdx.x;
  const int it = b % (LL / 16);
  const int h  = (b / (LL / 16)) % HH;
  const int n  = b / ((LL / 16) * HH);
  const int i0 = it * 16;

  const int tid  = threadIdx.x;
  const int w    = tid >> 5;
  const int lane = tid & 31;
  const int lm   = lane & 15;
  const int half = lane >> 4;

  // ---- phase 0: per-row geometric data --------------------------------
  if (tid < 16) {
    const int tk = n * LL + (i0 + tid);
    const float* Rm = R + (size_t)tk * 9;
    float* si = Si[tid];
    // Rinv = R^T (rotations)
    si[0] = Rm[0]; si[1] = Rm[3]; si[2] = Rm[6];
    si[3] = Rm[1]; si[4] = Rm[4]; si[5] = Rm[7];
    si[6] = Rm[2]; si[7] = Rm[5]; si[8] = Rm[8];
    const float* bp = bg + ((size_t)tk * HH + h) * 3;
    si[9] = bp[0]; si[10] = bp[1]; si[11] = bp[2];
    si[12] = t[(size_t)tk * 3]; si[13] = t[(size_t)tk * 3 + 1]; si[14] = t[(size_t)tk * 3 + 2];
    const float* ap = a + ((size_t)tk * HH + h) * (NQKc * 3);
#pragma unroll
    for (int vi = 0; vi < NQKc; ++vi) {
      const float ax = ap[vi * 3], ay = ap[vi * 3 + 1], az = ap[vi * 3 + 2];
      si[15 + vi * 3] = ax; si[16 + vi * 3] = ay; si[17 + vi * 3] = az;
      si[27 + vi] = sqrtf(ax * ax + ay * ay + az * az);
    }
  }
  __syncthreads();

  // ---- phase 1: scalar QK^T attention via WMMA ------------------------
  {
    const float* qrow = q + ((size_t)(n * LL + i0 + lm) * HH + h) * DQKc;
    for (int jt = w; jt < LL / 16; jt += 8) {
      const int j0 = jt * 16;
      const float* krow = k + ((size_t)(n * LL + j0 + lm) * HH + h) * DQKc;
      v8f acc = {};
#pragma unroll
      for (int kk = 0; kk < DQKc; kk += 4) {
        const int ka = kk + 2 * half;
        v2f af, bf;
        af.x = qrow[ka];  af.y = qrow[ka + 1];
        bf.x = krow[ka];  bf.y = krow[ka + 1];
        acc = wmma_f32x4(af, bf, acc);
      }
#pragma unroll
      for (int r = 0; r < 8; ++r)
        S[r + 8 * half][j0 + lm] = acc[r] * 0.25f;  // DQK^-0.5
    }
  }
  __syncthreads();

  // ---- phase 2: ray attention term + diag/mask ------------------------
  {
    const float sa = log1pf(expf(alpha[h]));   // softplus
    const float sb = log1pf(expf(beta[h]));
    for (int idx = tid; idx < 16 * LL; idx += 256) {
      const int il = idx / LL;
      const int j  = idx - il * LL;
      const float* si = Si[il];
      const float* kp = kvg + ((size_t)(n * LL + j) * HH + h) * (NQKc * 3);
      float rsum = 0.0f, tsum = 0.0f;
#pragma unroll
      for (int vi = 0; vi < NQKc; ++vi) {
        const float dx = kp[vi * 3]     - si[9];
        const float dy = kp[vi * 3 + 1] - si[10];
        const float dz = kp[vi * 3 + 2] - si[11];
        const float rx = si[0] * dx + si[1] * dy + si[2] * dz;
        const float ry = si[3] * dx + si[4] * dy + si[5] * dz;
        const float rz = si[6] * dx + si[7] * dy + si[8] * dz;
        const float rs = sqrtf(rx * rx + ry * ry + rz * rz);
        const float rda = rx * si[15 + vi * 3] + ry * si[16 + vi * 3] + rz * si[17 + vi * 3];
        float ct = rda / ((rs + EPSc) * (si[27 + vi] + EPSc));
        ct = fminf(1.0f, fmaxf(-1.0f, ct));
        rsum += rs;
        tsum += acosf(ct);
      }
      const float ray = -(sa * rsum + sb * tsum) * 0.5f;     // NQK^-0.5
      float val = 0.70710678118654752f * (ray + S[il][j]);
      if ((i0 + il) == j) val = -100.0f;
      if (!mask[n * LL + j]) val = -__builtin_inff();
      S[il][j] = val;
    }
  }
  __syncthreads();

  // ---- phase 3: softmax over j (row-wise, wave32 reductions) ----------
  for (int il = w; il < 16; il += 8) {
    float m = -__builtin_inff();
    for (int j = lane; j < LL; j += 32) m = fmaxf(m, S[il][j]);
#pragma unroll
    for (int off = 16; off; off >>= 1) m = fmaxf(m, __shfl_xor(m, off, 32));
    float sum = 0.0f;
    for (int j = lane; j < LL; j += 32) {
      const float e = expf(S[il][j] - m);
      S[il][j] = e;
      sum += e;
    }
#pragma unroll
    for (int off = 16; off; off >>= 1) sum += __shfl_xor(sum, off, 32);
    const float inv = 1.0f / sum;
    for (int j = lane; j < LL; j += 32) S[il][j] *= inv;
  }
  __syncthreads();

  // ---- phase 4: [o | ov_g] = att @ [v | vv_g] via WMMA ----------------
  // Pad lanes (cols 40..47) read a valid dummy address and scale by zero:
  // keeps the wave EXEC-uniform (no branches around loads, WMMA legal).
  if (w < 3) {
    const int col = w * 16 + lm;                 // 0..47 (40..47 padded)
    const float* vb;
    int vstride;
    float vscale = 1.0f;
    if (col < DVc) {
      vb = v + (size_t)n * LL * HH * DVc + h * DVc + col;
      vstride = HH * DVc;                        // per-j stride
    } else if (col < DVc + NVVc * 3) {
      const int cc = col - DVc;
      vb = vvg + (size_t)n * LL * HH * NVVc * 3 + (h * NVVc + cc / 3) * 3 + (cc % 3);
      vstride = HH * NVVc * 3;
    } else {
      vb = v + (size_t)n * LL * HH * DVc + h * DVc;   // valid dummy
      vstride = HH * DVc;
      vscale = 0.0f;
    }
    v8f acc = {};
    for (int k0 = 0; k0 < LL; k0 += 4) {
      const int ka = k0 + 2 * half;
      v2f af, bf;
      af.x = S[lm][ka];
      af.y = S[lm][ka + 1];
      bf.x = vb[(size_t)(ka)     * vstride] * vscale;
      bf.y = vb[(size_t)(ka + 1) * vstride] * vscale;
      acc = wmma_f32x4(af, bf, acc);
    }
#pragma unroll
    for (int r = 0; r < 8; ++r) Ot[r + 8 * half][col] = acc[r];
  }
  __syncthreads();

  // ---- phase 5: epilogue into feats -----------------------------------
  // feats per token: [0,192) o ; [192,480) ov(H*NVV,3) ; [480,576) |ov|
  {
    const int idx = tid;                          // 256 threads, 256 o-elems
    const int il = idx >> 4, c = idx & 15;
    feats[(size_t)(n * LL + i0 + il) * FEATc + h * DVc + c] = Ot[il][c];
  }
  for (int idx = tid; idx < 16 * NVVc; idx += 256) {
    const int il = idx >> 3, vi = idx & 7;
    const float* si = Si[il];
    const float gx = Ot[il][DVc + vi * 3]     - si[12];
    const float gy = Ot[il][DVc + vi * 3 + 1] - si[13];
    const float gz = Ot[il][DVc + vi * 3 + 2] - si[14];
    const float ox = si[0] * gx + si[1] * gy + si[2] * gz;
    const float oy = si[3] * gx + si[4] * gy + si[5] * gz;
    const float oz = si[6] * gx + si[7] * gy + si[8] * gz;
    float* f = feats + (size_t)(n * LL + i0 + il) * FEATc;
    f[192 + (h * NVVc + vi) * 3]     = ox;
    f[192 + (h * NVVc + vi) * 3 + 1] = oy;
    f[192 + (h * NVVc + vi) * 3 + 2] = oz;
    f[480 + h * NVVc + vi] = sqrtf(ox * ox + oy * oy + oz * oz);
  }
}

// ---------------------------------------------------------------------------
extern "C" void kernel_launch(void* const* d_in, const int* in_sizes, int n_in,
                              void* d_out, int out_size, void* d_ws, size_t ws_size,
                              hipStream_t stream) {
  const float* x    = (const float*)d_in[0];
  const float* R    = (const float*)d_in[1];
  const float* t    = (const float*)d_in[2];
  const unsigned char* mask = (const unsigned char*)d_in[3];
  const float* Wq   = (const float*)d_in[4];
  const float* Wk   = (const float*)d_in[5];
  const float* Wv   = (const float*)d_in[6];
  const float* Wa   = (const float*)d_in[7];
  const float* ba   = (const float*)d_in[8];
  const float* Wb   = (const float*)d_in[9];
  const float* bb   = (const float*)d_in[10];
  const float* Wkv  = (const float*)d_in[11];
  const float* Wvv  = (const float*)d_in[12];
  const float* alpha= (const float*)d_in[13];
  const float* beta = (const float*)d_in[14];
  const float* Wo   = (const float*)d_in[15];
  const float* bo   = (const float*)d_in[16];

  const int M = NN * LL;                 // 768 tokens
  float* ws = (float*)d_ws;
  float* q    = ws;                      // 768*192
  float* k    = q    + (size_t)M * 192;
  float* v    = k    + (size_t)M * 192;
  float* a    = v    + (size_t)M * 192;  // 768*144
  float* bg   = a    + (size_t)M * 144;  // 768*36
  float* kvg  = bg   + (size_t)M * 36;   // 768*144
  float* vvg  = kvg  + (size_t)M * 144;  // 768*288
  float* feats= vvg  + (size_t)M * 288;  // 768*576

  const dim3 blk(32);
  const int mt = M / 16;                 // 48

  // Projections (fp32 WMMA GEMMs)
  nra_gemm_bias_kernel<<<dim3(12, mt), blk, 0, stream>>>(x, Wq,  nullptr, q,   M, 192, HIDc);
  nra_gemm_bias_kernel<<<dim3(12, mt), blk, 0, stream>>>(x, Wk,  nullptr, k,   M, 192, HIDc);
  nra_gemm_bias_kernel<<<dim3(12, mt), blk, 0, stream>>>(x, Wv,  nullptr, v,   M, 192, HIDc);
  nra_gemm_bias_kernel<<<dim3( 9, mt), blk, 0, stream>>>(x, Wa,  ba,      a,   M, 144, HIDc);
  nra_gemm_bias_kernel<<<dim3( 3, mt), blk, 0, stream>>>(x, Wb,  bb,      bg,  M,  36, HIDc);
  nra_gemm_bias_kernel<<<dim3( 9, mt), blk, 0, stream>>>(x, Wkv, nullptr, kvg, M, 144, HIDc);
  nra_gemm_bias_kernel<<<dim3(18, mt), blk, 0, stream>>>(x, Wvv, nullptr, vvg, M, 288, HIDc);

  // Global-frame transform of b, kv, vv (in place)
  {
    const int tot = NN * LL * HH * (1 + NQKc + NVVc);   // 119808 vectors
    nra_frame_kernel<<<(tot + 255) / 256, 256, 0, stream>>>(R, t, bg, kvg, vvg);
  }

  // Fused attention (per n, h, 16-row i-tile)
  nra_attn_kernel<<<NN * HH * (LL / 16), 256, 0, stream>>>(
      q, k, v, a, bg, kvg, vvg, R, t, mask, alpha, beta, feats);

  // Output projection: out = feats @ Wo + bo
  nra_gemm_bias_kernel<<<dim3(HIDc / 16, mt), blk, 0, stream>>>(
      feats, Wo, bo, (float*)d_out, M, HIDc, FEATc);
}